// LocalAttentionLayer_42709154791515
// MI455X (gfx1250) — compile-verified
//
#include <hip/hip_runtime.h>
#include <math.h>

// Problem constants (match reference)
#define NB    32
#define NS    4096
#define DIN   1024
#define DOUT  1024

typedef __attribute__((ext_vector_type(2))) float v2f;
typedef __attribute__((ext_vector_type(8))) float v8f;

// ---------------------------------------------------------------------------
// Kernel 1/2: C[M,N] = A[M,K] * W[N,K]^T  via V_WMMA_F32_16X16X4_F32.
// One wave (32 lanes) per 16x16 C tile. Fragment layouts per CDNA5 ISA 7.12.2:
//   A (16x4 f32, 2 VGPRs/lane): lanes 0-15 -> M=lane, K={k,k+1};
//                               lanes 16-31 -> M=lane-16, K={k+2,k+3}
//   B (4x16  f32, 2 VGPRs/lane): lanes 0-15 -> N=lane, K={k,k+1};
//                               lanes 16-31 -> N=lane-16, K={k+2,k+3}
//   C (16x16 f32, 8 VGPRs): vgpr r -> row m0+r (lanes 0-15) / m0+r+8 (16-31)
// ---------------------------------------------------------------------------
__global__ void wmma_gemm_f32(const float* __restrict__ A,
                              const float* __restrict__ W,
                              float* __restrict__ C, int K, int N) {
    const int lane = threadIdx.x & 31;
    const int l16  = lane & 15;
    const int half = lane >> 4;            // 0: lanes 0-15, 1: lanes 16-31
    const int n0   = blockIdx.x * 16;
    const int m0   = blockIdx.y * 16;

    const float* arow = A + (size_t)(m0 + l16) * K + half * 2;
    const float* wrow = W + (size_t)(n0 + l16) * K + half * 2;

    v8f c = {};
    for (int k = 0; k < K; k += 4) {
        v2f a, b;
        a[0] = arow[k];  a[1] = arow[k + 1];
        b[0] = wrow[k];  b[1] = wrow[k + 1];
        // 8 args: (neg_a, A, neg_b, B, c_mod, C, reuse_a, reuse_b)
        c = __builtin_amdgcn_wmma_f32_16x16x4_f32(false, a, false, b,
                                                  (short)0, c, false, false);
    }
#pragma unroll
    for (int r = 0; r < 8; ++r) {
        const int row = m0 + r + half * 8;
        C[(size_t)row * N + n0 + l16] = c[r];
    }
}

// ---------------------------------------------------------------------------
// Kernel 3: mean_pos[b] = sigmoid( tanh(h_pre[b,:]+b1) . W2 + b2 ) * S
// ---------------------------------------------------------------------------
__global__ void meanpos_kernel(const float* __restrict__ hpre,
                               const float* __restrict__ b1,
                               const float* __restrict__ W2,
                               const float* __restrict__ b2,
                               float* __restrict__ mp) {
    const int b = blockIdx.x;
    __shared__ float red[256];
    float p = 0.f;
    for (int d = threadIdx.x; d < DIN; d += 256)
        p += tanhf(hpre[(size_t)b * DIN + d] + b1[d]) * W2[d];
    red[threadIdx.x] = p;
    __syncthreads();
    for (int s = 128; s > 0; s >>= 1) {
        if (threadIdx.x < s) red[threadIdx.x] += red[threadIdx.x + s];
        __syncthreads();
    }
    if (threadIdx.x == 0) {
        const float t = red[0] + b2[0];
        mp[b] = (1.f / (1.f + expf(-t))) * (float)NS;
    }
}

// ---------------------------------------------------------------------------
// Kernel 4 (THE hot pass, streams 512 MB): scores[b,s] = <hids[b,s,:], x[b,:]>
// 256 threads = 8 waves; each wave owns one s-row; x[b] cached in LDS.
// ---------------------------------------------------------------------------
__global__ void scores_kernel(const float* __restrict__ hids,
                              const float* __restrict__ x,
                              float* __restrict__ scores) {
    const int b = blockIdx.y;
    __shared__ float4 xs4[DOUT / 4];
    xs4[threadIdx.x] = ((const float4*)(x + (size_t)b * DOUT))[threadIdx.x];
    __syncthreads();

    const int wave = threadIdx.x >> 5;
    const int lane = threadIdx.x & 31;
    const int s    = blockIdx.x * 8 + wave;

    const float4* row = (const float4*)(hids + ((size_t)b * NS + s) * DOUT);
    float acc = 0.f;
#pragma unroll
    for (int j = 0; j < 8; ++j) {
        const int d4 = lane + j * 32;        // coalesced 512B per wave per j
        const float4 h  = row[d4];
        const float4 xv = xs4[d4];
        acc = fmaf(h.x, xv.x, acc);
        acc = fmaf(h.y, xv.y, acc);
        acc = fmaf(h.z, xv.z, acc);
        acc = fmaf(h.w, xv.w, acc);
    }
#pragma unroll
    for (int off = 16; off > 0; off >>= 1)
        acc += __shfl_xor(acc, off, 32);
    if (lane == 0) scores[(size_t)b * NS + s] = acc;
}

// ---------------------------------------------------------------------------
// Kernel 5: per-batch softmax statistics (max, sum of exp) over all S=4096.
// ---------------------------------------------------------------------------
__global__ void softmax_stats(const float* __restrict__ scores,
                              float* __restrict__ rmax,
                              float* __restrict__ rsum) {
    const int b = blockIdx.x;
    __shared__ float red[256];
    const float* sc = scores + (size_t)b * NS;

    float m = -INFINITY;
    for (int s = threadIdx.x; s < NS; s += 256) m = fmaxf(m, sc[s]);
    red[threadIdx.x] = m;
    __syncthreads();
    for (int st = 128; st > 0; st >>= 1) {
        if (threadIdx.x < st)
            red[threadIdx.x] = fmaxf(red[threadIdx.x], red[threadIdx.x + st]);
        __syncthreads();
    }
    m = red[0];
    __syncthreads();

    float sum = 0.f;
    for (int s = threadIdx.x; s < NS; s += 256) sum += expf(sc[s] - m);
    red[threadIdx.x] = sum;
    __syncthreads();
    for (int st = 128; st > 0; st >>= 1) {
        if (threadIdx.x < st) red[threadIdx.x] += red[threadIdx.x + st];
        __syncthreads();
    }
    if (threadIdx.x == 0) { rmax[b] = m; rsum[b] = red[0]; }
}

// ---------------------------------------------------------------------------
// Kernel 6: out[b,:] = sum_s softmax(s)*gauss(s) * hids[b,s,:]
// gauss = exp(-(s-mp)^2/2) underflows fp32 to exact 0 for |s-mp| > ~13.5
// (reference gets exact zeros there too), so only a +/-40 window matters.
// ---------------------------------------------------------------------------
__global__ void output_kernel(const float* __restrict__ hids,
                              const float* __restrict__ scores,
                              const float* __restrict__ mp,
                              const float* __restrict__ rmax,
                              const float* __restrict__ rsum,
                              float* __restrict__ out) {
    const int b = blockIdx.x;
    const float c   = mp[b];
    const float m   = rmax[b];
    const float inv = 1.f / rsum[b];

    int s0 = (int)floorf(c) - 40; if (s0 < 0)  s0 = 0;
    int s1 = (int)floorf(c) + 41; if (s1 > NS) s1 = NS;

    const int d4 = threadIdx.x;  // each of 256 threads owns one float4 of D
    float4 acc = make_float4(0.f, 0.f, 0.f, 0.f);
    const float4* base = (const float4*)(hids + (size_t)b * NS * DOUT);
    const float* sc = scores + (size_t)b * NS;

    for (int s = s0; s < s1; ++s) {
        const float dist = (float)s - c;
        const float w = expf(sc[s] - m) * inv * expf(-0.5f * dist * dist);
        const float4 h = base[(size_t)s * (DOUT / 4) + d4];
        acc.x = fmaf(w, h.x, acc.x);
        acc.y = fmaf(w, h.y, acc.y);
        acc.z = fmaf(w, h.z, acc.z);
        acc.w = fmaf(w, h.w, acc.w);
    }
    ((float4*)(out + (size_t)b * DOUT))[d4] = acc;
}

// ---------------------------------------------------------------------------
extern "C" void kernel_launch(void* const* d_in, const int* in_sizes, int n_in,
                              void* d_out, int out_size, void* d_ws, size_t ws_size,
                              hipStream_t stream) {
    const float* input = (const float*)d_in[0];   // [B, DIN]
    const float* hids  = (const float*)d_in[1];   // [B, S, DOUT]
    const float* Wp    = (const float*)d_in[2];   // [DOUT, DIN]
    const float* W1    = (const float*)d_in[3];   // [DIN, DIN]
    const float* b1    = (const float*)d_in[4];   // [DIN]
    const float* W2    = (const float*)d_in[5];   // [1, DIN]
    const float* b2    = (const float*)d_in[6];   // [1]
    float* out = (float*)d_out;

    // workspace layout (floats): x | hpre | scores | mp | rmax | rsum
    float* ws     = (float*)d_ws;
    float* x      = ws;                       // NB*DOUT
    float* hpre   = x + (size_t)NB * DOUT;    // NB*DIN
    float* scores = hpre + (size_t)NB * DIN;  // NB*NS
    float* mp     = scores + (size_t)NB * NS; // NB
    float* rmax   = mp + NB;                  // NB
    float* rsum   = rmax + NB;                // NB

    // 1) x = input @ Wp^T    (WMMA f32 16x16x4)
    wmma_gemm_f32<<<dim3(DOUT / 16, NB / 16), 32, 0, stream>>>(input, Wp, x, DIN, DOUT);
    // 2) hpre = input @ W1^T (WMMA f32 16x16x4)
    wmma_gemm_f32<<<dim3(DIN / 16, NB / 16), 32, 0, stream>>>(input, W1, hpre, DIN, DIN);
    // 3) mean_pos
    meanpos_kernel<<<NB, 256, 0, stream>>>(hpre, b1, W2, b2, mp);
    // 4) scores (streams the 512MB tensor once — the roofline-limiting pass)
    scores_kernel<<<dim3(NS / 8, NB), 256, 0, stream>>>(hids, x, scores);
    // 5) softmax stats
    softmax_stats<<<NB, 256, 0, stream>>>(scores, rmax, rsum);
    // 6) windowed weighted sum -> out
    output_kernel<<<NB, 256, 0, stream>>>(hids, scores, mp, rmax, rsum, out);
}